// GAT_86990267613313
// MI455X (gfx1250) — compile-verified
//
#include <hip/hip_runtime.h>
#include <hip/hip_bf16.h>
#include <math.h>

typedef float v2f __attribute__((ext_vector_type(2)));
typedef float v8f __attribute__((ext_vector_type(8)));

// ---------------------------------------------------------------- utilities
__global__ void fill_kernel(float* __restrict__ p, float v, int n) {
    int i = blockIdx.x * blockDim.x + threadIdx.x;
    if (i < n) p[i] = v;
}

__device__ __forceinline__ void atomicMaxF(float* addr, float v) {
    // sign-aware float max via integer atomics (monotone bit encodings)
    if (v >= 0.0f) atomicMax((int*)addr, __float_as_int(v));
    else           atomicMin((unsigned int*)addr, (unsigned int)__float_as_int(v));
}

// ---------------------------------------------------------------- WMMA GEMM
// C[M,N] = A[M,K] * B[K,N], fp32, one wave per 16x16 C tile.
// M multiple of 16, N multiple of 16, K (template) multiple of 4.
// A frag (16x4 f32): lane (k>=2?16:0)+m holds K=k in vgpr k&1.
// B frag (4x16 f32): lane (k>=2?16:0)+n holds K=k in vgpr k&1.
// C/D (16x16 f32): vgpr r, lanes 0-15 -> M=r, lanes 16-31 -> M=8+r.
template <int K>
__global__ __launch_bounds__(256) void wmma_gemm_f32(const float* __restrict__ A,
                                                     const float* __restrict__ B,
                                                     float* __restrict__ C,
                                                     int M, int N) {
    const int lane = threadIdx.x & 31;
    const int wave = blockIdx.x * (blockDim.x >> 5) + (threadIdx.x >> 5);
    const int ntiles = N >> 4;
    const int mt = wave / ntiles;
    const int nt = wave - mt * ntiles;
    if (mt * 16 >= M) return;               // whole-wave uniform exit (EXEC all-1 inside)

    const int hl  = lane >> 4;              // 0: K={0,1}, 1: K={2,3}
    const int l15 = lane & 15;

    const float* Arow = A + (size_t)(mt * 16 + l15) * K + 2 * hl;
    const float* Bcol = B + (size_t)(2 * hl) * N + nt * 16 + l15;

    v8f acc = {0.f, 0.f, 0.f, 0.f, 0.f, 0.f, 0.f, 0.f};
#pragma unroll
    for (int k0 = 0; k0 < K; k0 += 4) {
        v2f a, b;
        a.x = Arow[k0];
        a.y = Arow[k0 + 1];
        b.x = Bcol[(size_t)k0 * N];
        b.y = Bcol[(size_t)(k0 + 1) * N];
        acc = __builtin_amdgcn_wmma_f32_16x16x4_f32(false, a, false, b,
                                                    (short)0, acc, false, false);
    }

    float* Crow = C + (size_t)(mt * 16 + hl * 8) * N + nt * 16 + l15;
#pragma unroll
    for (int r = 0; r < 8; ++r) Crow[(size_t)r * N] = acc[r];
}

// ---------------------------------------------------------------- edge phase
// One thread per (edge, head): logit + atomic segment-max. float4 gathers.
__global__ void edge_logits_max(const float* __restrict__ xl, const float* __restrict__ xr,
                                const int* __restrict__ ei, const float* __restrict__ att,
                                float* __restrict__ logits, float* __restrict__ mbuf,
                                int E0, int Etot, int H, int C) {
    int t = blockIdx.x * blockDim.x + threadIdx.x;
    if (t >= Etot * H) return;
    int e = t / H, h = t - e * H;
    int s, d;
    if (e < E0) { s = ei[e]; d = ei[E0 + e]; } else { s = d = e - E0; }
    const int HC = H * C;
    const float4* pl = (const float4*)(xl + (size_t)s * HC + h * C);
    const float4* pr = (const float4*)(xr + (size_t)d * HC + h * C);
    const float4* pa = (const float4*)(att + h * C);
    float acc = 0.f;
    const int C4 = C >> 2;
    for (int c = 0; c < C4; ++c) {
        float4 l = pl[c], r = pr[c], a = pa[c];
        float v0 = l.x + r.x, v1 = l.y + r.y, v2 = l.z + r.z, v3 = l.w + r.w;
        v0 = (v0 > 0.f) ? v0 : 0.2f * v0;   // leaky_relu, NEG_SLOPE = 0.2
        v1 = (v1 > 0.f) ? v1 : 0.2f * v1;
        v2 = (v2 > 0.f) ? v2 : 0.2f * v2;
        v3 = (v3 > 0.f) ? v3 : 0.2f * v3;
        acc += a.x * v0 + a.y * v1 + a.z * v2 + a.w * v3;
    }
    logits[t] = acc;
    atomicMaxF(&mbuf[d * H + h], acc);
}

// One thread per (edge, head): ex = exp(logit - m[dst]); atomic segment-sum.
__global__ void edge_exp_sum(const int* __restrict__ ei, const float* __restrict__ mbuf,
                             float* __restrict__ logits /* in: logit, out: ex */,
                             float* __restrict__ sbuf, int E0, int Etot, int H) {
    int t = blockIdx.x * blockDim.x + threadIdx.x;
    if (t >= Etot * H) return;
    int e = t / H, h = t - e * H;
    int d = (e < E0) ? ei[E0 + e] : (e - E0);
    float ex = expf(logits[t] - mbuf[d * H + h]);
    logits[t] = ex;
    atomicAdd(&sbuf[d * H + h], ex);
}

// One thread per (edge, 4 channels): agg[dst] += xl[src] * alpha (float4 gather).
__global__ void edge_scatter(const float* __restrict__ xl, const int* __restrict__ ei,
                             const float* __restrict__ exb, const float* __restrict__ sbuf,
                             float* __restrict__ agg, int E0, int Etot, int H, int C) {
    const int HC = H * C;
    const int Q = HC >> 2;                  // float4 groups per edge
    long long t = (long long)blockIdx.x * blockDim.x + threadIdx.x;
    if (t >= (long long)Etot * Q) return;
    int e = (int)(t / Q);
    int q = (int)(t - (long long)e * Q);
    int j = q << 2;                         // first channel of this group
    int h = j / C;
    int s, d;
    if (e < E0) { s = ei[e]; d = ei[E0 + e]; } else { s = d = e - E0; }
    float alpha = exb[e * H + h] / (sbuf[d * H + h] + 1e-16f);
    float4 m = *(const float4*)(xl + (size_t)s * HC + j);
    float* ap = agg + (size_t)d * HC + j;
    atomicAdd(ap + 0, m.x * alpha);
    atomicAdd(ap + 1, m.y * alpha);
    atomicAdd(ap + 2, m.z * alpha);
    atomicAdd(ap + 3, m.w * alpha);
}

// x[i] = relu(x[i] + b[i % F]) in place.
__global__ void bias_relu(float* __restrict__ x, const float* __restrict__ b, int n, int F) {
    int i = blockIdx.x * blockDim.x + threadIdx.x;
    if (i >= n) return;
    float v = x[i] + b[i % F];
    x[i] = (v > 0.f) ? v : 0.f;
}

// ---------------------------------------------------------------- launch
static inline int cdiv_ll(long long a, long long b) { return (int)((a + b - 1) / b); }

extern "C" void kernel_launch(void* const* d_in, const int* in_sizes, int n_in,
                              void* d_out, int out_size, void* d_ws, size_t ws_size,
                              hipStream_t stream) {
    const float* x    = (const float*)d_in[0];
    const int*   ei   = (const int*)d_in[1];
    const float* Wl1  = (const float*)d_in[2];
    const float* Wr1  = (const float*)d_in[3];
    const float* att1 = (const float*)d_in[4];
    const float* b1   = (const float*)d_in[5];
    const float* Wl2  = (const float*)d_in[6];
    const float* Wr2  = (const float*)d_in[7];
    const float* att2 = (const float*)d_in[8];
    const float* b2   = (const float*)d_in[9];
    float* out = (float*)d_out;

    const int N  = in_sizes[0] / 128;   // 50000 (multiple of 16)
    const int E0 = in_sizes[1] / 2;     // 800000
    const int E  = E0 + N;              // + self loops
    const int H1 = 2, C1 = 64, HC1 = 128;
    const int C2 = 64;

    float* ws = (float*)d_ws;
    float* xl = ws;                          // N*128 (layer2 reuses N*64)
    float* xr = xl + (size_t)N * 128;        // N*128
    float* h1 = xr + (size_t)N * 128;        // N*128: agg1 then layer-1 activations
    float* eb = h1 + (size_t)N * 128;        // E*2 : logits -> ex
    float* mb = eb + (size_t)E * 2;          // N*2 : segment max
    float* sb = mb + (size_t)N * 2;          // N*2 : segment sum

    const int TPB = 256;

    // ================= layer 1 (H=2, C=64, concat) =================
    fill_kernel<<<cdiv_ll((long long)N * 128, TPB), TPB, 0, stream>>>(h1, 0.f, N * 128);
    fill_kernel<<<cdiv_ll(N * 2, TPB), TPB, 0, stream>>>(mb, -INFINITY, N * 2);
    fill_kernel<<<cdiv_ll(N * 2, TPB), TPB, 0, stream>>>(sb, 0.f, N * 2);

    {
        long long waves = (long long)(N / 16) * (128 / 16);
        int blocks = cdiv_ll(waves, 8);
        wmma_gemm_f32<128><<<blocks, TPB, 0, stream>>>(x, Wl1, xl, N, 128);
        wmma_gemm_f32<128><<<blocks, TPB, 0, stream>>>(x, Wr1, xr, N, 128);
    }

    edge_logits_max<<<cdiv_ll((long long)E * H1, TPB), TPB, 0, stream>>>(
        xl, xr, ei, att1, eb, mb, E0, E, H1, C1);
    edge_exp_sum<<<cdiv_ll((long long)E * H1, TPB), TPB, 0, stream>>>(
        ei, mb, eb, sb, E0, E, H1);
    edge_scatter<<<cdiv_ll((long long)E * (HC1 / 4), TPB), TPB, 0, stream>>>(
        xl, ei, eb, sb, h1, E0, E, H1, C1);
    bias_relu<<<cdiv_ll((long long)N * 128, TPB), TPB, 0, stream>>>(h1, b1, N * 128, 128);

    // ================= layer 2 (H=1, C=64, mean == identity) =================
    fill_kernel<<<cdiv_ll((long long)N * 64, TPB), TPB, 0, stream>>>(out, 0.f, N * 64);
    fill_kernel<<<cdiv_ll(N, TPB), TPB, 0, stream>>>(mb, -INFINITY, N);
    fill_kernel<<<cdiv_ll(N, TPB), TPB, 0, stream>>>(sb, 0.f, N);

    {
        long long waves = (long long)(N / 16) * (64 / 16);
        int blocks = cdiv_ll(waves, 8);
        wmma_gemm_f32<128><<<blocks, TPB, 0, stream>>>(h1, Wl2, xl, N, 64);
        wmma_gemm_f32<128><<<blocks, TPB, 0, stream>>>(h1, Wr2, xr, N, 64);
    }

    edge_logits_max<<<cdiv_ll((long long)E, TPB), TPB, 0, stream>>>(
        xl, xr, ei, att2, eb, mb, E0, E, 1, C2);
    edge_exp_sum<<<cdiv_ll((long long)E, TPB), TPB, 0, stream>>>(
        ei, mb, eb, sb, E0, E, 1);
    edge_scatter<<<cdiv_ll((long long)E * (C2 / 4), TPB), TPB, 0, stream>>>(
        xl, ei, eb, sb, out, E0, E, 1, C2);
    bias_relu<<<cdiv_ll((long long)N * 64, TPB), TPB, 0, stream>>>(out, b2, N * 64, 64);
}